// SNN1_6176162971648
// MI455X (gfx1250) — compile-verified
//
#include <hip/hip_runtime.h>
#include <stdint.h>

// ---------------------------------------------------------------------------
// LICell scan: v,i recurrence over T=500 steps, 131072 independent lanes.
// Memory-bound: 262 MB in + 262 MB out -> ~22.5 us floor @ 23.3 TB/s.
// Strategy: async-to-LDS prefetch FIFO (DEPTH time steps deep, ASYNCcnt
// tracked) feeding a register-resident scan; NT b128 stores for the output.
// ---------------------------------------------------------------------------

#define T_STEPS   500
#define PER_STEP  (32 * 64 * 64)          // 131072 floats per time step
#define VEC       4                       // floats per thread (b128)
#define BLOCK     128                     // 4 waves (wave32)
#define GRID      (PER_STEP / (VEC * BLOCK))  // 256 blocks
#define DEPTH     32                      // prefetch FIFO depth (time steps)
// LDS: DEPTH * BLOCK * VEC * 4B = 64 KB of the 320 KB WGP pool.
// In flight: GRID * DEPTH * (BLOCK*VEC*4B) = 16 MB > ~12 MB latency-BW product.

typedef float v4f __attribute__((ext_vector_type(4)));
typedef int   v4i __attribute__((vector_size(16)));
typedef v4i __attribute__((address_space(1)))* gp_v4i;   // global b128 pointer
typedef v4i __attribute__((address_space(3)))* lp_v4i;   // LDS b128 pointer

#if defined(__has_builtin)
#if __has_builtin(__builtin_amdgcn_global_load_async_to_lds_b128)
#define USE_ASYNC_BUILTIN 1
#endif
#endif

__device__ __forceinline__ void async_copy_b128(const float* gsrc, float* ldst) {
#ifdef USE_ASYNC_BUILTIN
  __builtin_amdgcn_global_load_async_to_lds_b128(
      (gp_v4i)(float*)gsrc, (lp_v4i)ldst, /*imm_offset=*/0, /*cpol=*/0);
#else
  // GV mode: vdst = LDS byte address VGPR, vaddr = 64-bit global address.
  uint32_t lofs  = (uint32_t)(size_t)(float __attribute__((address_space(3)))*)ldst;
  uint64_t gaddr = (uint64_t)(size_t)gsrc;
  asm volatile("global_load_async_to_lds_b128 %0, %1, off"
               :: "v"(lofs), "v"(gaddr) : "memory");
#endif
}

// DEPTH-1 == 31; keep literal in sync with DEPTH.
#define WAIT_ASYNC_OLDEST() asm volatile("s_wait_asynccnt 31" ::: "memory")
#define WAIT_ASYNC_ALL()    asm volatile("s_wait_asynccnt 0"  ::: "memory")
#define WAIT_DS_DRAIN()     asm volatile("s_wait_dscnt 0"     ::: "memory")

__global__ void __launch_bounds__(BLOCK)
li_scan_kernel(const float* __restrict__ x, float* __restrict__ out) {
  __shared__ float lds[DEPTH * BLOCK * VEC];

  const int     tid  = threadIdx.x;
  const int64_t base = (int64_t)blockIdx.x * (BLOCK * VEC) + tid * VEC;
  float* lslot = &lds[tid * VEC];   // slot s lives at lslot + s*BLOCK*VEC

  // Prime the FIFO with the first DEPTH time steps (no VGPR data cost).
#pragma unroll
  for (int p = 0; p < DEPTH; ++p) {
    async_copy_b128(x + (int64_t)p * PER_STEP + base, lslot + p * (BLOCK * VEC));
  }

  v4f v   = {0.f, 0.f, 0.f, 0.f};
  v4f cur = {0.f, 0.f, 0.f, 0.f};
  const float a = 0.5f;   // DT * TAU_MEM_INV
  const float d = 0.8f;   // 1 - DT * TAU_SYN_INV

  int t = 0;
  for (; t < T_STEPS - DEPTH; ++t) {
    WAIT_ASYNC_OLDEST();                       // step t's tile has landed in LDS
    const int slot = t & (DEPTH - 1);
    v4f xv = *(const v4f*)(lslot + slot * (BLOCK * VEC));   // ds_load_b128
    WAIT_DS_DRAIN();                           // slot consumed -> safe to refill
    async_copy_b128(x + (int64_t)(t + DEPTH) * PER_STEP + base,
                    lslot + slot * (BLOCK * VEC));

    cur = cur + xv;                            // i_new = i + x_t
    v   = v + a * (cur - v);                   // v_new
    __builtin_nontemporal_store(v, (v4f*)(out + (int64_t)t * PER_STEP + base));
    cur = cur * d;                             // i_dec
  }

  WAIT_ASYNC_ALL();                            // remaining DEPTH tiles resident
  for (; t < T_STEPS; ++t) {
    const int slot = t & (DEPTH - 1);
    v4f xv = *(const v4f*)(lslot + slot * (BLOCK * VEC));
    cur = cur + xv;
    v   = v + a * (cur - v);
    __builtin_nontemporal_store(v, (v4f*)(out + (int64_t)t * PER_STEP + base));
    cur = cur * d;
  }
}

extern "C" void kernel_launch(void* const* d_in, const int* in_sizes, int n_in,
                              void* d_out, int out_size, void* d_ws, size_t ws_size,
                              hipStream_t stream) {
  const float* x   = (const float*)d_in[0];
  float*       out = (float*)d_out;
  li_scan_kernel<<<GRID, BLOCK, 0, stream>>>(x, out);
}